// SimpleRegressionModel_22084721836458
// MI455X (gfx1250) — compile-verified
//
#include <hip/hip_runtime.h>
#include <hip/hip_bf16.h>
#include <cstdint>

// ---------------------------------------------------------------------------
// Presence-bag logistic regression:
//   logit[b] = sum_{unique id in token_ids[b,:]} weight[id] + bias
//   out[b]   = sigmoid(logit[b])
// Memory-latency bound (~2 MB total traffic @ 23.3 TB/s). One wave per row,
// TDM (tensor_load_to_lds) stages 8 rows of token ids per workgroup into LDS.
// ---------------------------------------------------------------------------

#define NTOK            200
#define ROWS_PER_BLOCK  8
#define TILE_ELEMS      (NTOK * ROWS_PER_BLOCK)   // 1600 ints = 6400 bytes

typedef uint32_t u32;
typedef unsigned int v4u __attribute__((ext_vector_type(4)));
typedef int          v4i __attribute__((ext_vector_type(4)));
typedef int          v8i __attribute__((ext_vector_type(8)));

__global__ __launch_bounds__(256) void presence_sigmoid_kernel(
    const int* __restrict__ token_ids,   // [batch, NTOK]
    const float* __restrict__ weight,    // [1, VOCAB]
    const float* __restrict__ bias,      // [1]
    float* __restrict__ out,             // [batch]
    int batch)
{
    __shared__ int ids[TILE_ELEMS];

    const int tid     = threadIdx.x;
    const int wave    = tid >> 5;
    const int lane    = tid & 31;
    const int rowBase = blockIdx.x * ROWS_PER_BLOCK;

#if defined(__gfx1250__) && __has_builtin(__builtin_amdgcn_tensor_load_to_lds)
    // --- Tensor Data Mover: DMA 6400 B of token ids (8 rows) into LDS. ---
    if (wave == 0) {
        const uint64_t gaddr   = (uint64_t)(uintptr_t)(token_ids + (size_t)rowBase * NTOK);
        const u32      ldsAddr = (u32)(uintptr_t)(&ids[0]);   // as3 ptr -> LDS byte offset

        // D# group 0 (128b): count=1 | lds_addr | global_addr[56:0] | type=2
        v4u g0;
        g0[0] = 1u;                                            // count=1, user desc, no gather
        g0[1] = ldsAddr;                                       // lds_addr
        g0[2] = (u32)(gaddr & 0xFFFFFFFFu);                    // global_addr[31:0]
        g0[3] = (u32)((gaddr >> 32) & 0x01FFFFFFu)             // global_addr[56:32]
              | (2u << 30);                                    // type = 2 ("image")

        // D# group 1 (256b): 1-D tile of 1600 x 4-byte elements
        v8i g1;
        g1[0] = (int)(2u << 16);                               // wg_mask=0, data_size=2 (4B)
        g1[1] = (int)((u32)(TILE_ELEMS & 0xFFFF) << 16);       // tensor_dim0[15:0]
        g1[2] = (int)(((u32)TILE_ELEMS >> 16) | (1u << 16));   // tensor_dim0[31:16], tensor_dim1=1
        g1[3] = (int)((u32)TILE_ELEMS << 16);                  // tile_dim0 = 1600
        g1[4] = 1;                                             // tile_dim1=1, tile_dim2=0
        g1[5] = TILE_ELEMS;                                    // tensor_dim0_stride[31:0]
        g1[6] = 0;                                             // stride0 hi, stride1 lo
        g1[7] = 0;                                             // stride1 hi

        v4i g2 = {0, 0, 0, 0};                                 // dims 2/3 unused
        v4i g3 = {0, 0, 0, 0};                                 // dim 4 unused
        v8i g4 = {0, 0, 0, 0, 0, 0, 0, 0};                     // extra group (unused, zero)

        __builtin_amdgcn_tensor_load_to_lds(g0, g1, g2, g3, g4, 0);
#if __has_builtin(__builtin_amdgcn_s_wait_tensorcnt)
        __builtin_amdgcn_s_wait_tensorcnt(0);
#else
        asm volatile("s_wait_tensorcnt 0" ::: "memory");
#endif
    }
    __syncthreads();   // publish DMA'd LDS tile to waves 1..7
#else
    // Fallback: cooperative vector loads into LDS.
    const size_t total = (size_t)batch * NTOK;
    for (int t = tid; t < TILE_ELEMS; t += 256) {
        size_t g = (size_t)rowBase * NTOK + t;
        ids[t] = (g < total) ? token_ids[g] : -1;
    }
    __syncthreads();
#endif

    const int row = rowBase + wave;
    if (row < batch) {
        const int* myIds = &ids[wave * NTOK];

        // Each lane handles positions t = lane, lane+32, ... ; a token counts
        // only if it is the FIRST occurrence in the row (scatter-set semantics).
        float acc = 0.0f;
        for (int t = lane; t < NTOK; t += 32) {
            const int id = myIds[t];
            bool first = true;
            for (int k = 0; k < t; ++k) {
                if (myIds[k] == id) { first = false; break; }
            }
            if (first) acc += weight[id];   // random 4B gather, L2-resident
        }

        // wave32 reduction
        for (int off = 16; off > 0; off >>= 1)
            acc += __shfl_down(acc, off, 32);

        if (lane == 0) {
            const float logit = acc + bias[0];
            out[row] = 1.0f / (1.0f + __expf(-logit));
        }
    }
}

extern "C" void kernel_launch(void* const* d_in, const int* in_sizes, int n_in,
                              void* d_out, int out_size, void* d_ws, size_t ws_size,
                              hipStream_t stream) {
    (void)n_in; (void)out_size; (void)d_ws; (void)ws_size;

    const int*   token_ids = (const int*)d_in[0];
    const float* weight    = (const float*)d_in[1];
    const float* bias      = (const float*)d_in[2];
    float*       out       = (float*)d_out;

    const int batch  = in_sizes[0] / NTOK;                       // 2048
    const int blocks = (batch + ROWS_PER_BLOCK - 1) / ROWS_PER_BLOCK;  // 256

    presence_sigmoid_kernel<<<blocks, 256, 0, stream>>>(token_ids, weight, bias, out, batch);
}